// MixtureSageLayer_4166118277411
// MI455X (gfx1250) — compile-verified
//
#include <hip/hip_runtime.h>

typedef float v2f __attribute__((ext_vector_type(2)));
typedef float v4f __attribute__((ext_vector_type(4)));
typedef float v8f __attribute__((ext_vector_type(8)));

#define N_NODES   100000
#define N_EDGES   1000000
#define IN_DIM    64
#define OUT_DIM   64
#define N_EXPERTS 8
#define BLOCK_M   64
#define LDS_STRIDE 132   // 128 + 4 pad: row stride 132 dwords spreads banks (4*i mod 64)

// packed-B geometry: [expert][nt=4][ktp=16][lane=32][4 floats]
#define PB_PER_EXPERT (4 * 16 * 32 * 4)          // 8192 floats
#define PB_TOTAL      (N_EXPERTS * PB_PER_EXPERT) // 65536 floats = 256 KB

// ---------------- kernel 1: zero workspace (sums + counts) ----------------
__global__ void zero_ws_kernel(float* __restrict__ ws, int n) {
    int i = blockIdx.x * blockDim.x + threadIdx.x;
    int stride = gridDim.x * blockDim.x;
    for (; i < n; i += stride) ws[i] = 0.0f;
}

// ---------------- kernel 1b: pack weights into WMMA B-fragment order ------
// pb[((e*4+nt)*16+ktp)*32+lane] (float4) = { w[kk][n], w[kk+1][n], w[kk+4][n], w[kk+5][n] }
// with kk = ktp*8 + (lane>>4)*2, n = nt*16 + (lane&15)
__global__ void pack_b_kernel(const float* __restrict__ w, float* __restrict__ pb) {
    int idx = blockIdx.x * blockDim.x + threadIdx.x;   // one float4 per thread
    if (idx >= N_EXPERTS * 4 * 16 * 32) return;
    int lane = idx & 31;
    int ktp  = (idx >> 5) & 15;
    int nt   = (idx >> 9) & 3;
    int e    = idx >> 11;
    int half = lane >> 4;
    int lrow = lane & 15;
    int kk = ktp * 8 + half * 2;
    int n  = nt * 16 + lrow;
    const float* we = w + e * 128 * 64;
    float4 v;
    v.x = we[(kk + 0) * 64 + n];
    v.y = we[(kk + 1) * 64 + n];
    v.z = we[(kk + 4) * 64 + n];
    v.w = we[(kk + 5) * 64 + n];
    *(float4*)(pb + (size_t)idx * 4) = v;
}

// ---------------- kernel 2: scatter-add (L2-resident fp32 atomics) --------
// work item = (edge, 16B chunk); 16 chunks of 4 floats per edge
__global__ void scatter_kernel(const float* __restrict__ x,
                               const int* __restrict__ edge,
                               float* __restrict__ sums,
                               float* __restrict__ cnt) {
    const long long total  = (long long)N_EDGES * 16;
    long long i      = (long long)blockIdx.x * blockDim.x + threadIdx.x;
    const long long stride = (long long)gridDim.x * blockDim.x;
    for (; i < total; i += stride) {
        int e = (int)(i >> 4);
        int c = (int)(i & 15);
        int dst = edge[e];              // row (destination)
        int src = edge[N_EDGES + e];    // col (source / gathered)
        const float4 v = *(const float4*)(x + (long long)src * IN_DIM + c * 4);
        float* s = sums + (long long)dst * IN_DIM + c * 4;
        atomicAdd(s + 0, v.x);
        atomicAdd(s + 1, v.y);
        atomicAdd(s + 2, v.z);
        atomicAdd(s + 3, v.w);
        if (c == 0) atomicAdd(cnt + dst, 1.0f);
    }
}

// ---------------- kernel 3: combined build + 8-expert GEMM via fp32 WMMA --
// 64 nodes/block, 8 waves, wave w owns expert w: 4x4 tiles of 16x16 f32,
// K=128 via 32 x v_wmma_f32_16x16x4_f32 (2 per packed-B b128 load)
__global__ __launch_bounds__(256, 1)
void moe_gemm_kernel(const float* __restrict__ x,
                     const float* __restrict__ sums,
                     const float* __restrict__ cnt,
                     const float* __restrict__ pb,
                     float* __restrict__ out) {
    __shared__ float comb[BLOCK_M * LDS_STRIDE];   // [64][132] fp32 ~ 33 KB

    const int base = blockIdx.x * BLOCK_M;
    const int tid  = threadIdx.x;

    // ---- stage combined = [agg | x] into LDS (zero-pad tail nodes) ----
    for (int idx = tid; idx < BLOCK_M * 128; idx += 256) {
        int m = idx >> 7;         // node within block
        int j = idx & 127;        // feature 0..127
        int node = base + m;
        float v = 0.0f;
        if (node < N_NODES) {
            if (j < 64) {
                float c = cnt[node];
                v = sums[(long long)node * 64 + j] / fmaxf(c, 1.0f);
            } else {
                v = x[(long long)node * 64 + (j - 64)];
            }
        }
        comb[m * LDS_STRIDE + j] = v;
    }
    __syncthreads();

    const int expert = tid >> 5;      // wave id == expert id (8 waves)
    const int lane   = tid & 31;
    const int half   = lane >> 4;     // 0: lanes 0-15, 1: lanes 16-31
    const int lrow   = lane & 15;
    const float* pbe = pb + (size_t)expert * PB_PER_EXPERT + (size_t)lane * 4;

    v8f acc[16] = {};                 // [nt*4 + mt], 128 VGPRs of accum

    for (int ktp = 0; ktp < 16; ++ktp) {   // K-step pairs: kt = 2*ktp, 2*ktp+1
        const int kk = ktp * 8 + half * 2; // per-lane K base for kt = 2*ktp

        // A fragments for 4 M-tiles, both K-steps (ds_load_2addr_b64 pairs)
        v2f a0[4], a1[4];
#pragma unroll
        for (int mt = 0; mt < 4; ++mt) {
            const float* p = &comb[(mt * 16 + lrow) * LDS_STRIDE + kk];
            a0[mt] = *(const v2f*)(p);       // K = kk, kk+1
            a1[mt] = *(const v2f*)(p + 4);   // K = kk+4, kk+5 (next kt)
        }

#pragma unroll
        for (int nt = 0; nt < 4; ++nt) {
            // one coalesced b128: both K-step B fragments for this lane
            v4f b = *(const v4f*)(pbe + (size_t)(nt * 16 + ktp) * 32 * 4);
            v2f b0 = __builtin_shufflevector(b, b, 0, 1);
            v2f b1 = __builtin_shufflevector(b, b, 2, 3);
#pragma unroll
            for (int mt = 0; mt < 4; ++mt) {
                acc[nt * 4 + mt] = __builtin_amdgcn_wmma_f32_16x16x4_f32(
                    false, a0[mt], false, b0, (short)0, acc[nt * 4 + mt], false, false);
            }
#pragma unroll
            for (int mt = 0; mt < 4; ++mt) {
                acc[nt * 4 + mt] = __builtin_amdgcn_wmma_f32_16x16x4_f32(
                    false, a1[mt], false, b1, (short)0, acc[nt * 4 + mt], false, false);
            }
        }
    }

    // ---- store D + residual (residual x read back from LDS combined) ----
#pragma unroll
    for (int nt = 0; nt < 4; ++nt) {
        const int o = nt * 16 + lrow;             // output feature
#pragma unroll
        for (int mt = 0; mt < 4; ++mt) {
            v8f d = acc[nt * 4 + mt];
#pragma unroll
            for (int r = 0; r < 8; ++r) {
                const int mloc = mt * 16 + half * 8 + r;   // C/D layout: M = r (+8 upper half)
                const int node = base + mloc;
                if (node < N_NODES) {
                    const float res = comb[mloc * LDS_STRIDE + 64 + o]; // = x[node][o]
                    out[((long long)node * N_EXPERTS + expert) * OUT_DIM + o] = d[r] + res;
                }
            }
        }
    }
}

// --------------------------------------------------------------------------
extern "C" void kernel_launch(void* const* d_in, const int* in_sizes, int n_in,
                              void* d_out, int out_size, void* d_ws, size_t ws_size,
                              hipStream_t stream) {
    (void)in_sizes; (void)n_in; (void)out_size; (void)ws_size;

    const float* x    = (const float*)d_in[0];   // [N, 64] fp32
    const int*   edge = (const int*)d_in[1];     // [2, E] int
    const float* wts  = (const float*)d_in[2];   // [8, 128, 64] fp32
    float*       out  = (float*)d_out;           // [N, 8, 64] fp32

    float* sums = (float*)d_ws;                              // [N, 64]
    float* cnt  = sums + (size_t)N_NODES * IN_DIM;           // [N]
    float* pb   = cnt  + (size_t)N_NODES;                    // packed B, 256 KB (16-float aligned)

    const int zn = N_NODES * (IN_DIM + 1);
    zero_ws_kernel<<<(zn + 255) / 256, 256, 0, stream>>>(sums, zn);

    pack_b_kernel<<<(N_EXPERTS * 4 * 16 * 32 + 255) / 256, 256, 0, stream>>>(wts, pb);

    scatter_kernel<<<4096, 256, 0, stream>>>(x, edge, sums, cnt);

    const int nblocks = (N_NODES + BLOCK_M - 1) / BLOCK_M;   // 1563
    moe_gemm_kernel<<<nblocks, 256, 0, stream>>>(x, sums, cnt, pb, out);
}